// V4TopKRouter_9637906612572
// MI455X (gfx1250) — compile-verified
//
#include <hip/hip_runtime.h>
#include <hip/hip_bf16.h>

typedef __attribute__((ext_vector_type(2))) float v2f;
typedef __attribute__((ext_vector_type(8))) float v8f;

#define D      4096
#define E      64
#define TOPK   8
#define ROWS   64               // rows per workgroup -> 256 blocks
#define KC     128              // K-chunk staged in LDS
#define SA_LD  132              // 132 mod 64 banks = 4 -> conflict-free A b64 reads
#define SB2_LD 160              // floats per k-pair row; 160 mod 64 = 32 -> lane halves disjoint
#define SS_LD  65               // score buffer stride (aliases sA)
#define EPSV   1e-12f

__launch_bounds__(256, 2)
__global__ void v4_topk_router_kernel(const float* __restrict__ hidden,
                                      const float* __restrict__ gate_w,
                                      const float* __restrict__ expert_bias,
                                      float* __restrict__ out) {
    __shared__ float sA[ROWS * SA_LD];        // hidden tile (reused as score buffer)
    __shared__ float sB[(KC / 2) * SB2_LD];   // gate_w^T tile as K-pairs: {k,k+1} per expert
    __shared__ float sBias[E];

    const int tid  = threadIdx.x;
    const int lane = tid & 31;
    const int wave = tid >> 5;                 // 8 waves
    const int row0 = blockIdx.x * ROWS;

    // wave -> (M-tile, expert-half):  waves 0..3 -> experts 0..31, waves 4..7 -> 32..63
    const int m0    = (wave & 3) * 16;
    const int nbase = (wave >> 2) * 32;

    if (tid < E) sBias[tid] = expert_bias[tid];

    // WMMA f32 16x16x4 fragment addressing (ISA 7.12.2, wave32):
    //  A: lanes 0-15 M=lane K={0,1}; lanes 16-31 M=lane-16 K={2,3}
    //  B: lanes 0-15 N=lane K={0,1}; lanes 16-31 N=lane-16 K={2,3}
    //  C/D: vgpr r -> M=r (lanes<16) / M=r+8 (lanes>=16); N=lane&15
    const int halfsel = lane >> 4;
    const int ar      = m0 + (lane & 15);
    const int koff    = halfsel << 1;
    const int bcol    = lane & 15;

    v8f acc0 = {}, acc1 = {};

    for (int ko = 0; ko < D; ko += KC) {
        // ---- stage hidden tile via gfx1250 async global->LDS (ASYNCcnt path) ----
        #pragma unroll
        for (int i = 0; i < (ROWS * KC / 4) / 256; ++i) {   // 8 iters, B128 per thread
            int idx = tid + i * 256;
            int r   = idx >> 5;                  // row in tile (const per wave)
            int kq  = idx & 31;                  // float4 index along K
            unsigned ldsoff = (unsigned)(uintptr_t)&sA[r * SA_LD + kq * 4];
            unsigned goff   = (unsigned)(((row0 + r) * (size_t)D + ko + kq * 4) * 4u);
            asm volatile("global_load_async_to_lds_b128 %0, %1, %2"
                         :: "v"(ldsoff), "v"(goff), "s"(hidden) : "memory");
        }

        // ---- stage gate tile as swizzled K-pairs (overlaps with async copies) ----
        #pragma unroll
        for (int i = 0; i < (E * KC / 4) / 256; ++i) {      // 8 iters
            int idx = tid + i * 256;
            int e   = idx >> 5;
            int kq  = idx & 31;
            float4 v = *(const float4*)(gate_w + (size_t)e * D + ko + kq * 4);
            int kp0 = 2 * kq, kp1 = 2 * kq + 1;
            v2f p0 = {v.x, v.y};
            v2f p1 = {v.z, v.w};
            *(v2f*)&sB[kp0 * SB2_LD + ((2 * e) ^ ((kp0 & 31) << 1))] = p0;
            *(v2f*)&sB[kp1 * SB2_LD + ((2 * e) ^ ((kp1 & 31) << 1))] = p1;
        }
        if (ko + KC < D)   // keep next hidden chunk flowing into L2
            __builtin_prefetch(hidden + (size_t)(row0 + (tid & 63)) * D + ko + KC
                                       + ((tid >> 6) * (KC / 4)), 0, 2);

        asm volatile("s_wait_asynccnt 0x0" ::: "memory");
        __syncthreads();

        // ---- 32 K-quads of FP32 WMMA per chunk ----
        #pragma unroll 4
        for (int kk = 0; kk < KC; kk += 4) {
            v2f a = *(const v2f*)(&sA[ar * SA_LD + kk + koff]);        // ds_load_b64

            int kp = (kk >> 1) + halfsel;
            int sw = (kp & 31) << 1;
            const float* brow = &sB[kp * SB2_LD];
            v2f b0 = *(const v2f*)(&brow[(((nbase      + bcol) << 1) ^ sw)]);
            v2f b1 = *(const v2f*)(&brow[(((nbase + 16 + bcol) << 1) ^ sw)]);

            acc0 = __builtin_amdgcn_wmma_f32_16x16x4_f32(false, a, false, b0, (short)0, acc0, false, false);
            acc1 = __builtin_amdgcn_wmma_f32_16x16x4_f32(false, a, false, b1, (short)0, acc1, false, false);
        }
        __syncthreads();
    }

    // ---- epilogue: score = sqrt(softplus(logit)) -> LDS (aliases sA) ----
    float* sS = sA;                              // 64 x 64 (stride SS_LD) fits in sA
    const int crow = m0 + (halfsel ? 8 : 0);
    #pragma unroll
    for (int r = 0; r < 8; ++r) {
        float l0 = acc0[r], l1 = acc1[r];
        #define SQSP(x) sqrtf(fmaxf((x), 0.f) + log1pf(__expf(-fabsf(x))))
        sS[(crow + r) * SS_LD + (nbase      + bcol)] = SQSP(l0);
        sS[(crow + r) * SS_LD + (nbase + 16 + bcol)] = SQSP(l1);
        #undef SQSP
    }
    __syncthreads();

    // ---- top-8 per row: one thread per row ----
    if (tid < ROWS) {
        const int row = tid;
        float bv[TOPK];
        int   bi[TOPK];
        #pragma unroll
        for (int j = 0; j < TOPK; ++j) { bv[j] = -3.4e38f; bi[j] = 0; }

        #pragma unroll 8
        for (int e = 0; e < E; ++e) {
            float sel = sS[row * SS_LD + e] + sBias[e];
            if (sel > bv[TOPK - 1]) {
                bv[TOPK - 1] = sel; bi[TOPK - 1] = e;
                #pragma unroll
                for (int j = TOPK - 1; j > 0; --j) {
                    if (bv[j] > bv[j - 1]) {
                        float tv = bv[j]; bv[j] = bv[j - 1]; bv[j - 1] = tv;
                        int   ti = bi[j]; bi[j] = bi[j - 1]; bi[j - 1] = ti;
                    }
                }
            }
        }

        float sv[TOPK];
        float sum = 0.f;
        #pragma unroll
        for (int j = 0; j < TOPK; ++j) { sv[j] = sS[row * SS_LD + bi[j]]; sum += sv[j]; }
        float inv = 1.0f / fmaxf(sum, EPSV);

        const size_t N = 16384;
        float* probs = out;
        float* rmap  = out + N * E;              // routing_map emitted as 0.0/1.0
        const size_t base = (size_t)(row0 + row) * E;

        float4 z = {0.f, 0.f, 0.f, 0.f};
        #pragma unroll
        for (int e = 0; e < E; e += 4) {
            *(float4*)(probs + base + e) = z;
            *(float4*)(rmap  + base + e) = z;
        }
        #pragma unroll
        for (int j = 0; j < TOPK; ++j) {
            probs[base + bi[j]] = sv[j] * inv;
            rmap [base + bi[j]] = 1.0f;
        }
    }
}

extern "C" void kernel_launch(void* const* d_in, const int* in_sizes, int n_in,
                              void* d_out, int out_size, void* d_ws, size_t ws_size,
                              hipStream_t stream) {
    const float* hidden      = (const float*)d_in[0];   // [4,4096,4096] f32
    const float* gate_w      = (const float*)d_in[1];   // [64,4096] f32
    const float* expert_bias = (const float*)d_in[2];   // [64] f32
    float* out = (float*)d_out;                         // probs ++ routing_map

    const int N = 16384;
    dim3 grid(N / ROWS);      // 256 blocks
    dim3 block(256);          // 8 waves (wave32)
    v4_topk_router_kernel<<<grid, block, 0, stream>>>(hidden, gate_w, expert_bias, out);
}